// MatMulCorrelationBlock_45414984188032
// MI455X (gfx1250) — compile-verified
//
#include <hip/hip_runtime.h>
#include <hip/hip_bf16.h>

// Correlation layer: result[b, j*9+i, h1, w1] =
//   (1/256) * sum_c x1[b,c,2h1,2w1] * x2[b,c,2h1+2j-8,2w1+2i-8], 0 if OOB.
// One workgroup per (b, h1, j): 24x48x256 fp32 GEMM via V_WMMA_F32_16X16X4_F32.
// Staging of the two 48x128 fp32 tiles uses the Tensor Data Mover
// (tensor_load_to_lds + s_wait_tensorcnt) when the toolchain exposes it.

typedef __attribute__((ext_vector_type(2))) float v2f;
typedef __attribute__((ext_vector_type(8))) float v8f;
typedef __attribute__((ext_vector_type(4))) unsigned int u32x4;
typedef __attribute__((ext_vector_type(4))) int i32x4;
typedef __attribute__((ext_vector_type(8))) int i32x8;

#define CC   256   // channels (K)
#define HH   48
#define WW   48
#define H1   24
#define W1   24
#define ND   9     // displacement count per axis
#define KC   128   // K chunk
#define NTHREADS 192

#if defined(__AMDGCN__) && __has_builtin(__builtin_amdgcn_tensor_load_to_lds) && \
    __has_builtin(__builtin_amdgcn_s_wait_tensorcnt)
#define USE_TDM 1
#else
#define USE_TDM 0
#endif

#if USE_TDM
// Issue one TDM 2D tile load: tile_dim0 x tile_dim1 elements of 4 bytes,
// rows strided by strideElems in global memory, packed contiguously into LDS.
__device__ __forceinline__ void tdm_load_2d(unsigned int ldsByteOff,
                                            const float* gsrc,
                                            int tileD0, int tileD1,
                                            int strideElems)
{
    const unsigned long long ga = (unsigned long long)(uintptr_t)gsrc;

    u32x4 g0;
    g0.x = 1u;                                       // count=1, user descriptor
    g0.y = ldsByteOff;                               // D#.lds_addr (bytes)
    g0.z = (unsigned int)(ga & 0xFFFFFFFFu);         // global_addr[31:0]
    g0.w = (unsigned int)((ga >> 32) & 0x1FFFFFFu)   // global_addr[56:32]
         | (2u << 30);                               // type=2 ("image")

    i32x8 g1;
    g1[0] = (int)(2u << 16);                         // wg_mask=0, data_size=2 (4B)
    g1[1] = (int)((unsigned)tileD0 << 16);           // tensor_dim0[15:0] @ bits[63:48]
    g1[2] = (int)((unsigned)tileD1 << 16);           // tensor_dim1[15:0] @ bits[95:80]
    g1[3] = (int)((unsigned)tileD0 << 16);           // tile_dim0 @ bits[127:112]
    g1[4] = tileD1;                                  // tile_dim1 @ bits[143:128], tile_dim2=0
    g1[5] = strideElems;                             // tensor_dim0_stride[31:0]
    g1[6] = 0;                                       // stride hi / dim1_stride lo
    g1[7] = 0;

    i32x4 g2 = {0, 0, 0, 0};                         // 2D tensor: groups 2/3 unused
    i32x4 g3 = {0, 0, 0, 0};

#if __has_include(<hip/amd_detail/amd_gfx1250_TDM.h>)
    i32x8 g4 = {0, 0, 0, 0, 0, 0, 0, 0};             // therock-10.0 6-arg form
    __builtin_amdgcn_tensor_load_to_lds(g0, g1, g2, g3, g4, 0);
#else
    __builtin_amdgcn_tensor_load_to_lds(g0, g1, g2, g3, 0);   // ROCm 7.2 5-arg form
#endif
}
#endif

__global__ __launch_bounds__(NTHREADS)
void corr_wmma_kernel(const float* __restrict__ x1,
                      const float* __restrict__ x2,
                      float* __restrict__ out)
{
    const int jdy = blockIdx.x;   // y-displacement index 0..8
    const int h1  = blockIdx.y;   // 0..23
    const int b   = blockIdx.z;   // 0..15
    const int dy  = 2 * jdy - 8;
    const int y   = 2 * h1 + dy;
    const int tid = threadIdx.x;

    const size_t outBase = (size_t)b * (81 * H1 * W1) + (size_t)(jdy * ND) * (H1 * W1)
                         + (size_t)h1 * W1;

    // Entire key row out of range -> all 216 outputs for this block are zero.
    if (y < 0 || y >= HH) {
        for (int t = tid; t < ND * W1; t += NTHREADS) {
            const int i  = t / W1;
            const int w1 = t % W1;
            out[outBase + (size_t)i * (H1 * W1) + w1] = 0.0f;
        }
        return;
    }

    __shared__ float ldsA[KC * WW];   // x1 rows: [k][x], 48-wide (A[m][k] = ldsA[k*48+2m])
    __shared__ float ldsB[KC * WW];   // x2 rows: [k][x]
    __shared__ float ldsG[H1 * WW];   // G[w1][x] result tile

    // Wave -> output tile assignment: 6 waves = 2 M-tiles x 3 N-tiles.
    const int wave = tid >> 5;
    const int lane = tid & 31;
    const int lm   = lane & 15;
    const int half = lane >> 4;       // K-halves / M-halves per ISA 7.12.2 layouts
    const int m0   = (wave / 3) * 16;
    const int n0   = (wave % 3) * 16;

    const size_t x1RowBase = ((size_t)b * CC * HH + (size_t)(2 * h1)) * WW; // +c*HH*WW + x
    const size_t x2RowBase = ((size_t)b * CC * HH + (size_t)y) * WW;        // +c*HH*WW + x

    // Speculative prefetch of the second K-chunk (global_prefetch_b8).
    __builtin_prefetch(&x2[x2RowBase + (size_t)KC * HH * WW], 0, 0);
    __builtin_prefetch(&x1[x1RowBase + (size_t)KC * HH * WW], 0, 0);

    // Loop-invariant LDS columns for the operand fragments.
    int acol = 2 * (m0 + lm);
    if (acol >= WW) acol = 0;         // padded M rows (m>=24): harmless, discarded later
    const int bcol = n0 + lm;

    v8f acc = {};

    for (int kc = 0; kc < CC; kc += KC) {
#if USE_TDM
        if (wave == 0) {
            tdm_load_2d((unsigned int)(uintptr_t)&ldsA[0],
                        &x1[x1RowBase + (size_t)kc * (HH * WW)], WW, KC, HH * WW);
            tdm_load_2d((unsigned int)(uintptr_t)&ldsB[0],
                        &x2[x2RowBase + (size_t)kc * (HH * WW)], WW, KC, HH * WW);
            __builtin_amdgcn_s_wait_tensorcnt(0);   // DMA complete before release
        }
        __syncthreads();
#else
        for (int idx = tid; idx < KC * WW; idx += NTHREADS) {
            const int kk = idx / WW;
            const int x  = idx % WW;                // coalesced along x
            ldsA[idx] = x1[x1RowBase + (size_t)(kc + kk) * (HH * WW) + x];
            ldsB[idx] = x2[x2RowBase + (size_t)(kc + kk) * (HH * WW) + x];
        }
        __syncthreads();
#endif

        // ---- 32 WMMA K-steps over this chunk ----
        const float* aBase = &ldsA[(2 * half) * WW + acol];
        const float* bBase = &ldsB[(2 * half) * WW + bcol];
#pragma unroll 8
        for (int k = 0; k < KC; k += 4) {
            // A 16x4: lane<16 holds K={k,k+1}, lane>=16 holds K={k+2,k+3}
            v2f av = { aBase[k * WW], aBase[k * WW + WW] };
            // B 4x16 mirrored layout
            v2f bv = { bBase[k * WW], bBase[k * WW + WW] };
            acc = __builtin_amdgcn_wmma_f32_16x16x4_f32(
                      /*neg_a=*/false, av, /*neg_b=*/false, bv,
                      /*c_mod=*/(short)0, acc, /*reuse_a=*/false, /*reuse_b=*/false);
        }
        __syncthreads();
    }

    // ---- Spill accumulator tile to LDS (scaled by 1/C) ----
    union { v8f v; float f[8]; } u;
    u.v = acc;
#pragma unroll
    for (int r = 0; r < 8; ++r) {
        const int m = m0 + r + 8 * half;   // C/D layout: VGPR r -> M=r (lo) / r+8 (hi)
        if (m < H1)
            ldsG[m * WW + (n0 + lm)] = u.f[r] * (1.0f / 256.0f);
    }
    __syncthreads();

    // ---- Masked gather of the 9 x-displacements per query column ----
    for (int t = tid; t < ND * W1; t += NTHREADS) {
        const int i  = t / W1;             // x-displacement index
        const int w1 = t % W1;
        const int x  = 2 * w1 + 2 * i - 8;
        const float val = (x >= 0 && x < WW) ? ldsG[w1 * WW + x] : 0.0f;
        out[outBase + (size_t)i * (H1 * W1) + w1] = val;
    }
}

extern "C" void kernel_launch(void* const* d_in, const int* in_sizes, int n_in,
                              void* d_out, int out_size, void* d_ws, size_t ws_size,
                              hipStream_t stream)
{
    (void)in_sizes; (void)n_in; (void)d_ws; (void)ws_size; (void)out_size;
    const float* x1 = (const float*)d_in[0];
    const float* x2 = (const float*)d_in[1];
    float* out = (float*)d_out;

    dim3 grid(ND, H1, 16);        // (j, h1, b)
    dim3 block(NTHREADS);         // 6 wave32s: 2 M-tiles x 3 N-tiles
    corr_wmma_kernel<<<grid, block, 0, stream>>>(x1, x2, out);
}